// EMEncoder_21483426414987
// MI455X (gfx1250) — compile-verified
//
#include <hip/hip_runtime.h>

typedef __attribute__((ext_vector_type(2))) float v2f;
typedef __attribute__((ext_vector_type(8))) float v8f;

#define HDIM 512
#define NLAB 5
#define BIGF 1e11f

// Output layout (flat concat, reference return order), all float32:
//   tar_aug       [8][5][512]      off 0
//   tar_aug_mask  [8][5]           off 20480
//   ref_aug       [8][32][5][512]  off 20520
//   ref_aug_mask  [8][32][5]       off 675880
//   tarpaper      [8][512]         off 677160
//   tar_mask2     [8]              off 681256
//   refpaper      [8][32][512]     off 681264
//   ref_mask2     [8][32]          off 812336   (total 812592)

__global__ __launch_bounds__(256)
void emenc_wmma_kernel(const float* __restrict__ tar_state,
                       const int*   __restrict__ tar_func,
                       const float* __restrict__ ref_state,
                       const int*   __restrict__ ref_func,
                       float*       __restrict__ out)
{
    const int B = 8, TS = 512, RS = 256;

    float* tar_aug      = out;
    float* tar_aug_mask = out + 20480;
    float* ref_aug      = out + 20520;
    float* ref_aug_mask = out + 675880;
    float* tar_paper    = out + 677160;
    float* tar_mask2    = out + 681256;
    float* ref_paper    = out + 681264;
    float* ref_mask2    = out + 812336;

    __shared__ int lds_func[512];
    __shared__ int lds_cnt[8];

    const int doc = blockIdx.x;      // 0..7 tar, 8..263 ref (flat b*32+d)
    const float* state;
    const int*   func;
    int S;
    float *aug_out, *mask_out, *paper_out, *mask2_out;
    if (doc < B) {
        state     = tar_state + (size_t)doc * TS * HDIM;
        func      = tar_func  + doc * TS;
        S         = TS;
        aug_out   = tar_aug      + doc * NLAB * HDIM;
        mask_out  = tar_aug_mask + doc * NLAB;
        paper_out = tar_paper    + doc * HDIM;
        mask2_out = tar_mask2    + doc;
    } else {
        const int idx = doc - B;     // b*32 + d
        state     = ref_state + (size_t)idx * RS * HDIM;
        func      = ref_func  + idx * RS;
        S         = RS;
        aug_out   = ref_aug      + idx * NLAB * HDIM;
        mask_out  = ref_aug_mask + idx * NLAB;
        paper_out = ref_paper    + idx * HDIM;
        mask2_out = ref_mask2    + idx;
    }

    const int tid = threadIdx.x;

    // Stage func[] in LDS; zero label counters.
    for (int s = tid; s < S; s += 256) lds_func[s] = func[s];
    if (tid < 8) lds_cnt[tid] = 0;
    __syncthreads();

    // Label counts via LDS atomics (overlaps with WMMA loop issue).
    for (int s = tid; s < S; s += 256) {
        int f = lds_func[s];
        if (f > 0) {
            atomicAdd(&lds_cnt[f - 1], 1);
            atomicAdd(&lds_cnt[5], 1);   // non-pad total (paper denom)
        }
    }

    // ---- WMMA one-hot segment-sum: C[16,16] += A[16,4] (one-hot) x B[4,16] (state) ----
    const int wave = tid >> 5;
    const int lane = tid & 31;
    const int m    = lane & 15;          // A row / C column index within lane group
    const int kb   = (lane >> 4) << 1;   // lanes 0-15 -> K0/K1, lanes 16-31 -> K2/K3
    const int h0   = blockIdx.y * 128 + wave * 16;

    const float* colp = state + h0 + m;  // B column n == (lane & 15)

    // Branchless per-lane one-hot generator: bit f of matchMask says whether
    // label value f lights up this A-row.  Rows 0..4 match labels 1..5,
    // row 5 matches any non-pad (paper sum), rows 6..15 stay zero.
    unsigned matchMask = (m < 5) ? (2u << m) : ((m == 5) ? 0x3Eu : 0u);

    v8f acc = {};
#pragma unroll 4
    for (int s = 0; s < S; s += 4) {
        const unsigned f0 = (unsigned)lds_func[s + kb];
        const unsigned f1 = (unsigned)lds_func[s + kb + 1];
        v2f a, b;
        a.x = (float)((matchMask >> f0) & 1u);
        a.y = (float)((matchMask >> f1) & 1u);
        b.x = colp[(size_t)(s + kb)     * HDIM];
        b.y = colp[(size_t)(s + kb + 1) * HDIM];
        // D = A*B + C, exact f32 accumulation
        acc = __builtin_amdgcn_wmma_f32_16x16x4_f32(
                  /*neg_a=*/false, a, /*neg_b=*/false, b,
                  /*c_mod=*/(short)0, acc, /*reuse_a=*/false, /*reuse_b=*/false);
    }

    __syncthreads();  // counts final

    float den[6];
#pragma unroll
    for (int l = 0; l < 6; ++l) {
        const int c = lds_cnt[l];
        den[l] = (c > 0) ? (float)c : BIGF;
    }

    // C layout: VGPR r, lanes 0-15 hold row M=r, column N=lane.
    if (lane < 16) {
        const int h = h0 + lane;
        aug_out[0 * HDIM + h] = acc[0] / den[0];
        aug_out[1 * HDIM + h] = acc[1] / den[1];
        aug_out[2 * HDIM + h] = acc[2] / den[2];
        aug_out[3 * HDIM + h] = acc[3] / den[3];
        aug_out[4 * HDIM + h] = acc[4] / den[4];
        paper_out[h]          = acc[5] / den[5];
    }

    // Masks written once per doc (h-chunk 0 only).
    if (blockIdx.y == 0) {
        if (tid < 5)  mask_out[tid] = (lds_cnt[tid] > 0) ? 1.0f : 0.0f;
        if (tid == 5) *mask2_out    = (lds_cnt[5]  > 0) ? 1.0f : 0.0f;
    }
}

extern "C" void kernel_launch(void* const* d_in, const int* in_sizes, int n_in,
                              void* d_out, int out_size, void* d_ws, size_t ws_size,
                              hipStream_t stream) {
    (void)in_sizes; (void)n_in; (void)out_size; (void)d_ws; (void)ws_size;
    const float* tar_state = (const float*)d_in[0];   // [8,512,512] f32
    const int*   tar_func  = (const int*)  d_in[1];   // [8,512] i32
    const float* ref_state = (const float*)d_in[2];   // [8,32,256,512] f32
    const int*   ref_func  = (const int*)  d_in[3];   // [8,32,256] i32
    float* out = (float*)d_out;

    dim3 grid(8 + 8 * 32, 4, 1);   // 264 docs x 4 h-chunks of 128
    dim3 block(256, 1, 1);         // 8 wave32s, one 16-wide h tile each
    emenc_wmma_kernel<<<grid, block, 0, stream>>>(tar_state, tar_func,
                                                  ref_state, ref_func, out);
}